// PointTransformer_4234837753926
// MI455X (gfx1250) — compile-verified
//
#include <hip/hip_runtime.h>
#include <hip/hip_bf16.h>

// ---------------------------------------------------------------------------
// PointTransformer forward for MI455X (gfx1250, wave32).
// Dense algebra runs through V_WMMA_F32_16X16X4_F32 (exact fp32 semantics,
// matrix pipe). GEMM stages A (row-major) and B (column-major) chunks in LDS,
// zero-padded, so the inner K-loop is branchless wave-uniform ds_load_b64 +
// v_wmma only, with operand pairs loaded contiguously (no repack movs).
// ---------------------------------------------------------------------------

typedef __attribute__((ext_vector_type(2))) float v2f;
typedef __attribute__((ext_vector_type(8))) float v8f;

#define BATCH 2
#define NPTS  8192
#define KNN_K 16
#define EPS_BN 1e-5f

static __device__ __forceinline__ int imin(int a, int b) { return a < b ? a : b; }

// ---------------------------------------------------------------------------
// fp32 WMMA GEMM:  C[M,N] = A[M,K] @ B[K,N] (+bias) (+relu)
// grid = (ceil(M/32), ceil(N/128)), block = 256 (8 wave32s).
// Wave w: waveM = w&1 (16-row half), waveN = w>>1 (32-col group).
// Each wave: one A-frag shared across two 16x16 WMMA accumulators.
// LDS: A chunk 32x32 row-major (stride 36), B chunk 128x32 column-major
// (stride 34: rows 8B-aligned, per-lane b64 reads bank-conflict-free).
// Chunks are zero-padded -> inner loop has no guards, EXEC all-ones.
// ---------------------------------------------------------------------------
__global__ void k_gemm_wmma(const float* __restrict__ A, const float* __restrict__ B,
                            const float* __restrict__ bias, float* __restrict__ C,
                            int M, int N, int Kd, int relu)
{
    __shared__ float As[32 * 36];    // As[row][k]   row-major
    __shared__ float Bs[128 * 34];   // Bs[col][k]   column-major
    const int lane = threadIdx.x & 31;
    const int wave = threadIdx.x >> 5;
    const int waveM = wave & 1;
    const int waveN = wave >> 1;
    const int mr = lane & 15;   // row-in-tile for A, col-in-tile for B/C
    const int kh = lane >> 4;   // K-pair selector (0 -> K0/K1, 1 -> K2/K3)
    const int m0b = blockIdx.x * 32;          // block row base
    const int nb0 = blockIdx.y * 128;         // block col base
    const int arow = waveM * 16 + mr;         // this lane's A row (local)
    const int bcol = waveN * 32 + mr;         // this lane's B col (local)

    v8f acc0 = {};
    v8f acc1 = {};
    for (int kc = 0; kc < Kd; kc += 32) {
        const int kcw = imin(32, Kd - kc);
        // ---- cooperative zero-padded chunk loads (outside the hot loop) ----
        for (int t = threadIdx.x; t < 32 * 32; t += 256) {
            const int r = t >> 5, c = t & 31;
            float v = 0.0f;
            if (c < kcw && (m0b + r) < M) v = A[(size_t)(m0b + r) * Kd + kc + c];
            As[r * 36 + c] = v;
        }
        for (int t = threadIdx.x; t < 32 * 128; t += 256) {
            const int r = t >> 7, c = t & 127;   // r = k index, c = col
            float v = 0.0f;
            if (r < kcw && (nb0 + c) < N) v = B[(size_t)(kc + r) * N + nb0 + c];
            Bs[c * 34 + r] = v;                  // transpose into column-major
        }
        __syncthreads();
        // ---- branchless, wave-uniform WMMA loop (EXEC all-ones) ----
#pragma unroll
        for (int kk = 0; kk < 32; kk += 4) {
            const int k0 = kk + 2 * kh;
            v2f a, b0, b1;
            a.x  = As[arow * 36 + k0];
            a.y  = As[arow * 36 + k0 + 1];
            b0.x = Bs[bcol * 34 + k0];
            b0.y = Bs[bcol * 34 + k0 + 1];
            b1.x = Bs[(bcol + 16) * 34 + k0];
            b1.y = Bs[(bcol + 16) * 34 + k0 + 1];
            acc0 = __builtin_amdgcn_wmma_f32_16x16x4_f32(
                false, a, false, b0, (short)0, acc0, false, false);
            acc1 = __builtin_amdgcn_wmma_f32_16x16x4_f32(
                false, a, false, b1, (short)0, acc1, false, false);
        }
        __syncthreads();
    }
    // ---- masked epilogue ----
    const int col0 = nb0 + waveN * 32 + mr;
    const int col1 = col0 + 16;
    const float bv0 = (bias && col0 < N) ? bias[col0] : 0.0f;
    const float bv1 = (bias && col1 < N) ? bias[col1] : 0.0f;
#pragma unroll
    for (int r = 0; r < 8; ++r) {
        const int row = m0b + waveM * 16 + r + 8 * kh;
        if (row < M) {
            if (col0 < N) {
                float v = acc0[r] + bv0;
                if (relu) v = fmaxf(v, 0.0f);
                C[(size_t)row * N + col0] = v;
            }
            if (col1 < N) {
                float v = acc1[r] + bv1;
                if (relu) v = fmaxf(v, 0.0f);
                C[(size_t)row * N + col1] = v;
            }
        }
    }
}

// ---------------------------------------------------------------------------
// x0[r,c] = w[0,c]*1 + b[c]   (mlp_in applied to all-ones input)
// ---------------------------------------------------------------------------
__global__ void k_fill_x0(const float* __restrict__ w, const float* __restrict__ b,
                          float* __restrict__ x, int total, int C)
{
    int t = blockIdx.x * blockDim.x + threadIdx.x;
    if (t >= total) return;
    int c = t % C;
    x[t] = w[c] + b[c];
}

// ---------------------------------------------------------------------------
// Column-wise BatchNorm (training mode, biased var) + ReLU. One block per
// column c; in-place safe (y may alias x).
// ---------------------------------------------------------------------------
__global__ void k_bn_relu(const float* __restrict__ x, const float* __restrict__ g,
                          const float* __restrict__ be, float* __restrict__ y,
                          int rows, int C)
{
    const int c = blockIdx.x;
    __shared__ float s1[256], s2[256];
    float sum = 0.0f, sq = 0.0f;
    for (int r = threadIdx.x; r < rows; r += 256) {
        float v = x[(size_t)r * C + c];
        sum += v; sq += v * v;
    }
    s1[threadIdx.x] = sum; s2[threadIdx.x] = sq;
    __syncthreads();
    for (int off = 128; off > 0; off >>= 1) {
        if (threadIdx.x < off) {
            s1[threadIdx.x] += s1[threadIdx.x + off];
            s2[threadIdx.x] += s2[threadIdx.x + off];
        }
        __syncthreads();
    }
    const float inv_n = 1.0f / (float)rows;
    const float mu = s1[0] * inv_n;
    const float var = s2[0] * inv_n - mu * mu;
    const float sc = g[c] * rsqrtf(var + EPS_BN);
    const float sh = be[c];
    for (int r = threadIdx.x; r < rows; r += 256) {
        float v = (x[(size_t)r * C + c] - mu) * sc + sh;
        y[(size_t)r * C + c] = fmaxf(v, 0.0f);
    }
}

// ---------------------------------------------------------------------------
// kNN: for each query point, 16 nearest db points (sq L2), insertion select.
// exclude_self=1 for self-graph mode (db == query, skip j == qi).
// Tie-break: strict '<' keeps the lower index first (matches stable top_k).
// ---------------------------------------------------------------------------
__global__ void k_knn(const float* __restrict__ db, const float* __restrict__ q,
                      int* __restrict__ out, int n_db, int n_q, int nb,
                      int exclude_self)
{
    int t = blockIdx.x * blockDim.x + threadIdx.x;
    if (t >= nb * n_q) return;
    const int qi = t % n_q;
    const int b  = t / n_q;
    const float* dbb = db + (size_t)b * n_db * 3;
    const float qx = q[(size_t)t * 3 + 0];
    const float qy = q[(size_t)t * 3 + 1];
    const float qz = q[(size_t)t * 3 + 2];
    float bd[KNN_K];
    int   bi[KNN_K];
#pragma unroll
    for (int i = 0; i < KNN_K; ++i) { bd[i] = 3.4e38f; bi[i] = 0; }
    for (int j = 0; j < n_db; ++j) {
        if (exclude_self && j == qi) continue;
        const float dx = dbb[j * 3 + 0] - qx;
        const float dy = dbb[j * 3 + 1] - qy;
        const float dz = dbb[j * 3 + 2] - qz;
        const float d = dx * dx + dy * dy + dz * dz;
        if (d < bd[KNN_K - 1]) {
            int p = KNN_K - 1;
            while (p > 0 && bd[p - 1] > d) {
                bd[p] = bd[p - 1]; bi[p] = bi[p - 1]; --p;
            }
            bd[p] = d; bi[p] = j;
        }
    }
#pragma unroll
    for (int i = 0; i < KNN_K; ++i) out[(size_t)t * KNN_K + i] = bi[i];
}

// ---------------------------------------------------------------------------
// Farthest point sampling: one block per batch, dists live in LDS (n<=8192).
// Matches reference: idx[0]=0; argmax prefers the first (lowest) index.
// ---------------------------------------------------------------------------
__global__ void k_fps(const float* __restrict__ pos, int* __restrict__ idx,
                      int n, int m)
{
    const int b = blockIdx.x;
    const float* p = pos + (size_t)b * n * 3;
    int* id = idx + (size_t)b * m;
    __shared__ float dists[NPTS];
    __shared__ float rmx[256];
    __shared__ int   rmi[256];
    __shared__ int   s_last;
    for (int j = threadIdx.x; j < n; j += 256) dists[j] = 3.4e38f;
    if (threadIdx.x == 0) { id[0] = 0; s_last = 0; }
    __syncthreads();
    for (int i = 1; i < m; ++i) {
        const int last = s_last;
        const float lx = p[last * 3 + 0];
        const float ly = p[last * 3 + 1];
        const float lz = p[last * 3 + 2];
        float bmax = -1.0f; int bj = 0x7fffffff;
        for (int j = threadIdx.x; j < n; j += 256) {
            const float dx = p[j * 3 + 0] - lx;
            const float dy = p[j * 3 + 1] - ly;
            const float dz = p[j * 3 + 2] - lz;
            const float d = dx * dx + dy * dy + dz * dz;
            const float nd = fminf(dists[j], d);
            dists[j] = nd;
            if (nd > bmax) { bmax = nd; bj = j; }  // strided j ascending -> first max kept
        }
        rmx[threadIdx.x] = bmax; rmi[threadIdx.x] = bj;
        __syncthreads();
        for (int off = 128; off > 0; off >>= 1) {
            if (threadIdx.x < off) {
                const float ov = rmx[threadIdx.x + off];
                const int   oi = rmi[threadIdx.x + off];
                if (ov > rmx[threadIdx.x] ||
                    (ov == rmx[threadIdx.x] && oi < rmi[threadIdx.x])) {
                    rmx[threadIdx.x] = ov; rmi[threadIdx.x] = oi;
                }
            }
            __syncthreads();
        }
        if (threadIdx.x == 0) { id[i] = rmi[0]; s_last = rmi[0]; }
        __syncthreads();
    }
}

// sub[b,i,:] = pos[b, idx[b,i], :]
__global__ void k_gather_pos(const float* __restrict__ pos, const int* __restrict__ idx,
                             float* __restrict__ sub, int n, int m, int nb)
{
    int t = blockIdx.x * blockDim.x + threadIdx.x;
    if (t >= nb * m * 3) return;
    const int d = t % 3;
    const int bi = t / 3;
    const int b = bi / m;
    sub[t] = pos[((size_t)b * n + idx[bi]) * 3 + d];
}

// xout[b,i,c] = max_k h[b, nbr[b,i,k], c]
__global__ void k_gather_max(const float* __restrict__ h, const int* __restrict__ nbr,
                             float* __restrict__ xout, int n, int m, int C, int nb)
{
    int t = blockIdx.x * blockDim.x + threadIdx.x;
    if (t >= nb * m * C) return;
    const int c = t % C;
    const int node = t / C;          // b*m + i
    const int b = node / m;
    float best = -3.4e38f;
#pragma unroll
    for (int kk = 0; kk < KNN_K; ++kk) {
        const int j = nbr[(size_t)node * KNN_K + kk];
        best = fmaxf(best, h[((size_t)b * n + j) * C + c]);
    }
    xout[t] = best;
}

// rel[e,:] = pos_i - pos_j, edges e = node*17 + kk, kk==16 -> self loop
__global__ void k_edge_pos(const float* __restrict__ pos, const int* __restrict__ nbr,
                           float* __restrict__ rel, int Nl, int Etot)
{
    int e = blockIdx.x * blockDim.x + threadIdx.x;
    if (e >= Etot) return;
    const int kk = e % 17;
    const int node = e / 17;        // b*Nl + i
    const int i = node % Nl;
    const int b = node / Nl;
    const int j = (kk < KNN_K) ? nbr[(size_t)node * KNN_K + kk] : i;
    const float* pi = pos + (size_t)node * 3;
    const float* pj = pos + ((size_t)b * Nl + j) * 3;
    rel[(size_t)e * 3 + 0] = pi[0] - pj[0];
    rel[(size_t)e * 3 + 1] = pi[1] - pj[1];
    rel[(size_t)e * 3 + 2] = pi[2] - pj[2];
}

// ain[e,c] = a_dst[i,c] - a_src[j,c] + delta[e,c]
__global__ void k_edge_alpha(const float* __restrict__ adst, const float* __restrict__ asrc,
                             const float* __restrict__ delta, const int* __restrict__ nbr,
                             float* __restrict__ ain, int Nl, int C, int total)
{
    int t = blockIdx.x * blockDim.x + threadIdx.x;
    if (t >= total) return;
    const int c = t % C;
    const int e = t / C;
    const int kk = e % 17;
    const int node = e / 17;
    const int i = node % Nl;
    const int b = node / Nl;
    const int j = (kk < KNN_K) ? nbr[(size_t)node * KNN_K + kk] : i;
    ain[t] = adst[(size_t)node * C + c] - asrc[((size_t)b * Nl + j) * C + c] + delta[t];
}

// Per-(node,channel) softmax over the 17 incident edges, then weighted sum
// of (v_j + delta_e).
__global__ void k_softmax_agg(const float* __restrict__ alpha, const float* __restrict__ v,
                              const float* __restrict__ delta, const int* __restrict__ nbr,
                              float* __restrict__ agg, int Nl, int C, int total)
{
    int t = blockIdx.x * blockDim.x + threadIdx.x;
    if (t >= total) return;
    const int c = t % C;
    const int node = t / C;
    const int i = node % Nl;
    const int b = node / Nl;
    float av[17];
    float mx = -3.4e38f;
#pragma unroll
    for (int kk = 0; kk < 17; ++kk) {
        const float a = alpha[((size_t)node * 17 + kk) * C + c];
        av[kk] = a;
        mx = fmaxf(mx, a);
    }
    float s = 0.0f;
#pragma unroll
    for (int kk = 0; kk < 17; ++kk) { av[kk] = __expf(av[kk] - mx); s += av[kk]; }
    const float inv = 1.0f / s;
    float out = 0.0f;
#pragma unroll
    for (int kk = 0; kk < 17; ++kk) {
        const int j = (kk < KNN_K) ? nbr[(size_t)node * KNN_K + kk] : i;
        const size_t e = (size_t)node * 17 + kk;
        out += (av[kk] * inv) * (v[((size_t)b * Nl + j) * C + c] + delta[e * C + c]);
    }
    agg[t] = out;
}

// xg[b,c] = mean_i x[b,i,c]
__global__ void k_meanpool(const float* __restrict__ x, float* __restrict__ xg,
                           int m, int C, int nb)
{
    int t = blockIdx.x * blockDim.x + threadIdx.x;
    if (t >= nb * C) return;
    const int c = t % C;
    const int b = t / C;
    float s = 0.0f;
    for (int i = 0; i < m; ++i) s += x[((size_t)b * m + i) * C + c];
    xg[t] = s / (float)m;
}

// ---------------------------------------------------------------------------
// Host orchestration
// ---------------------------------------------------------------------------
struct TBP {
    const float *attn_w1, *attn_b1, *attn_w2, *attn_b2;
    const float *lin_in_w, *lin_in_b, *lin_out_w, *lin_out_b;
    const float *pos_w1, *pos_b1, *pos_w2, *pos_b2;
    const float *wdst, *wlin, *wsrc;
};

extern "C" void kernel_launch(void* const* d_in, const int* in_sizes, int n_in,
                              void* d_out, int out_size, void* d_ws, size_t ws_size,
                              hipStream_t stream)
{
    (void)in_sizes; (void)n_in; (void)out_size; (void)ws_size;
    static const int dims[5] = {32, 64, 128, 256, 512};

    // --- input parsing (JAX pytree order: dicts by sorted key, lists/tuples
    //     in order). Top: params, pos. params: lin, mlp_in, tb0, tbs, td.
    int cur = 0;
    auto nxt = [&]() { return (const float*)d_in[cur++]; };
    auto parse_tb = [&](TBP& p) {
        // keys sorted: attn_nn, lin_in, lin_out, pos_nn, wdst, wlin, wsrc
        p.attn_w1 = nxt(); p.attn_b1 = nxt(); p.attn_w2 = nxt(); p.attn_b2 = nxt();
        p.lin_in_w = nxt(); p.lin_in_b = nxt();
        p.lin_out_w = nxt(); p.lin_out_b = nxt();
        p.pos_w1 = nxt(); p.pos_b1 = nxt(); p.pos_w2 = nxt(); p.pos_b2 = nxt();
        p.wdst = nxt(); p.wlin = nxt(); p.wsrc = nxt();
    };

    // params["lin"]: be, g, (w, b)
    const float* fin_be = nxt(); const float* fin_g = nxt();
    const float* fin_w = nxt();  const float* fin_b = nxt();
    // params["mlp_in"]: be, g, (w, b)
    const float* min_be = nxt(); const float* min_g = nxt();
    const float* min_w = nxt();  const float* min_b = nxt();
    // params["tb0"]
    TBP tb0; parse_tb(tb0);
    // params["tbs"][0..3]
    TBP tbs[4]; for (int i = 0; i < 4; ++i) parse_tb(tbs[i]);
    // params["td"][0..3]: be, g, (w, b)
    struct TD { const float *be, *g, *w, *b; } td[4];
    for (int i = 0; i < 4; ++i) { td[i].be = nxt(); td[i].g = nxt(); td[i].w = nxt(); td[i].b = nxt(); }
    // pos
    const float* pos_in = nxt();

    // --- workspace bump allocator
    char* ws = (char*)d_ws;
    size_t off = 0;
    auto allocf = [&](size_t elems) -> float* {
        float* p = (float*)(ws + off);
        off += ((elems * sizeof(float)) + 255) & ~(size_t)255;
        return p;
    };
    auto alloci = [&](size_t elems) -> int* {
        int* p = (int*)(ws + off);
        off += ((elems * sizeof(int)) + 255) & ~(size_t)255;
        return p;
    };

    const size_t NODE_MAX = (size_t)BATCH * NPTS;         // 16384
    const size_t E_MAX = NODE_MAX * 17;                   // 278528
    float* XA   = allocf(NODE_MAX * 64);
    float* XB   = allocf(NODE_MAX * 64);
    float* H    = allocf(NODE_MAX * 64);
    float* X1   = allocf(NODE_MAX * 32);
    float* V    = allocf(NODE_MAX * 32);
    float* ASRC = allocf(NODE_MAX * 32);
    float* ADST = allocf(NODE_MAX * 32);
    float* AGG  = allocf(NODE_MAX * 32);
    float* REL  = allocf(E_MAX * 3);
    float* T64  = allocf(E_MAX * 64);
    float* DELTA = allocf(E_MAX * 32);
    float* AIN   = allocf(E_MAX * 32);
    float* ALPHA = allocf(E_MAX * 32);
    float* POSA = allocf((size_t)BATCH * NPTS * 3);
    float* POSB = allocf((size_t)BATCH * NPTS * 3);
    float* XG   = allocf((size_t)BATCH * 512);
    int* NBR   = alloci(NODE_MAX * KNN_K);
    int* NBRTD = alloci((size_t)BATCH * 2048 * KNN_K);
    int* IDX   = alloci((size_t)BATCH * 2048);

    auto gemm = [&](const float* A, const float* Bm, const float* bias, float* C,
                    int M, int N, int K, int relu) {
        dim3 g((M + 31) / 32, (N + 127) / 128);
        k_gemm_wmma<<<g, 256, 0, stream>>>(A, Bm, bias, C, M, N, K, relu);
    };

    auto run_tb = [&](const TBP& p, const float* pos, const int* nbr,
                      const float* xin, float* xout, int Nl, int C) {
        const int rows = BATCH * Nl;
        const int E = rows * 17;
        gemm(xin, p.lin_in_w, p.lin_in_b, X1, rows, C, C, 1);
        gemm(X1, p.wlin, nullptr, V,    rows, C, C, 0);
        gemm(X1, p.wsrc, nullptr, ASRC, rows, C, C, 0);
        gemm(X1, p.wdst, nullptr, ADST, rows, C, C, 0);
        k_edge_pos<<<(E + 255) / 256, 256, 0, stream>>>(pos, nbr, REL, Nl, E);
        gemm(REL, p.pos_w1, p.pos_b1, T64, E, 64, 3, 1);
        gemm(T64, p.pos_w2, p.pos_b2, DELTA, E, C, 64, 1);
        const int ec = E * C;
        k_edge_alpha<<<(ec + 255) / 256, 256, 0, stream>>>(ADST, ASRC, DELTA, nbr, AIN, Nl, C, ec);
        gemm(AIN, p.attn_w1, p.attn_b1, T64, E, 64, C, 1);
        gemm(T64, p.attn_w2, p.attn_b2, ALPHA, E, C, 64, 1);
        const int rc = rows * C;
        k_softmax_agg<<<(rc + 255) / 256, 256, 0, stream>>>(ALPHA, V, DELTA, nbr, AGG, Nl, C, rc);
        gemm(AGG, p.lin_out_w, p.lin_out_b, xout, rows, C, C, 1);
    };

    // ---- mlp_in: ones @ w + b, then BN+ReLU over all B*N rows
    {
        const int total = BATCH * NPTS * dims[0];
        k_fill_x0<<<(total + 255) / 256, 256, 0, stream>>>(min_w, min_b, XA, total, dims[0]);
        k_bn_relu<<<dims[0], 256, 0, stream>>>(XA, min_g, min_be, XA, BATCH * NPTS, dims[0]);
    }

    // ---- level-0 graph + transformer block
    const float* pos_cur = pos_in;
    int n = NPTS;
    k_knn<<<(BATCH * n + 63) / 64, 64, 0, stream>>>(pos_cur, pos_cur, NBR, n, n, BATCH, 1);
    run_tb(tb0, pos_cur, NBR, XA, XB, n, dims[0]);
    float* xcur = XB;
    float* xoth = XA;

    // ---- 4 downsampling levels
    for (int i = 0; i < 4; ++i) {
        const int m = (n + 3) / 4;
        const int Cin = dims[i], Cout = dims[i + 1];
        k_fps<<<BATCH, 256, 0, stream>>>(pos_cur, IDX, n, m);
        float* sub = (i % 2 == 0) ? POSA : POSB;
        k_gather_pos<<<(BATCH * m * 3 + 255) / 256, 256, 0, stream>>>(pos_cur, IDX, sub, n, m, BATCH);
        k_knn<<<(BATCH * m + 63) / 64, 64, 0, stream>>>(pos_cur, sub, NBRTD, n, m, BATCH, 0);
        // h = BN+ReLU( x @ w + b )
        gemm(xcur, td[i].w, td[i].b, H, BATCH * n, Cout, Cin, 0);
        k_bn_relu<<<Cout, 256, 0, stream>>>(H, td[i].g, td[i].be, H, BATCH * n, Cout);
        // pooled features via max over fixed-k clusters
        k_gather_max<<<(BATCH * m * Cout + 255) / 256, 256, 0, stream>>>(H, NBRTD, xoth, n, m, Cout, BATCH);
        pos_cur = sub;
        n = m;
        k_knn<<<(BATCH * n + 63) / 64, 64, 0, stream>>>(pos_cur, pos_cur, NBR, n, n, BATCH, 1);
        run_tb(tbs[i], pos_cur, NBR, xoth, xcur, n, Cout);
        // xcur holds current features; xoth free for next level
    }

    // ---- global mean pool + final linear + BN + ReLU
    k_meanpool<<<(BATCH * 512 + 255) / 256, 256, 0, stream>>>(xcur, XG, n, 512, BATCH);
    gemm(XG, fin_w, fin_b, H, BATCH, 512, 512, 0);
    k_bn_relu<<<512, 256, 0, stream>>>(H, fin_g, fin_be, (float*)d_out, BATCH, 512);
}